// ObjectDecoder_48498770706564
// MI455X (gfx1250) — compile-verified
//
#include <hip/hip_runtime.h>

// ObjectDecoder: out[b,o,a,p,k] = sum_d x[b,o,d] * W[o,a,p,d,k] + bias[o,a,p,k]
// == 16 independent GEMMs  X_o[16384x256] * B_o[256x64] + bias_o,
// B_o[d][n] = W[o][n>>3][d][n&7],  n = a*16+p*8+k in [0,64).
//
// fp32 -> V_WMMA_F32_16X16X4_F32 (2048 FLOP/inst).
// HBM floor ~14.4us (335 MB @ 23.3 TB/s). x streamed once -> NT loads;
// out written once -> NT stores; W kept L2/LDS-resident (RT loads).

typedef __attribute__((ext_vector_type(2))) float v2f;
typedef __attribute__((ext_vector_type(8))) float v8f;

#define BDIM            256      // 8 waves (wave32)
#define ROWS_PER_ITER   128      // 8 waves * 16 rows
#define NITER           4
#define ROWS_PER_BLOCK  (ROWS_PER_ITER * NITER)   // 512
#define N_TOT           64
#define K_TOT           256
#define NPAIRS          (K_TOT / 2)               // 128 k-pairs
#define PAIR_STRIDE     160      // dwords: 128 data + 32 pad; %64==32 ->
                                 // lane halves use disjoint bank halves

extern __shared__ float wlds[];  // NPAIRS * PAIR_STRIDE floats = 80 KB

__device__ __forceinline__ v2f nt_load2(const float* p) {
    return __builtin_nontemporal_load((const v2f*)p);
}

// One chunk = 4 K-steps (K=16) x 4 N-tiles = 16 WMMAs, A frags in registers.
__device__ __forceinline__ void chunk4(v8f (&acc)[4], const v2f (&a)[4],
                                       const float* __restrict__ wl0) {
#pragma unroll
    for (int j = 0; j < 4; ++j) {
        const float* __restrict__ wl = wl0 + (2 * j) * PAIR_STRIDE;
        v2f bfr[4];
#pragma unroll
        for (int t = 0; t < 4; ++t)           // issue both ds_load_2addr_b64
            bfr[t] = *(const v2f*)(wl + t * 32);
#pragma unroll
        for (int t = 0; t < 4; ++t)
            acc[t] = __builtin_amdgcn_wmma_f32_16x16x4_f32(
                false, a[j], false, bfr[t], (short)0, acc[t], false, false);
    }
}

__global__ __launch_bounds__(BDIM)
void objdec_wmma_f32(const float* __restrict__ x,
                     const float* __restrict__ W,
                     const float* __restrict__ bias,
                     float* __restrict__ out)
{
    const int o   = blockIdx.y;
    const int tid = threadIdx.x;
    const float* __restrict__ Wo = W + (size_t)o * (N_TOT * K_TOT);

    // ---- Stage B_o into LDS, interleaved by k-pair:
    //   wlds[p*PAIR_STRIDE + n*2 + j] = Wo[(n>>3)*2048 + (2p+j)*8 + (n&7)]
    for (int i = tid; i < NPAIRS * N_TOT; i += BDIM) {
        const int p = i >> 6;
        const int n = i & 63;
        const int gbase = (n >> 3) * 2048 + p * 16 + (n & 7);
        const float g0 = Wo[gbase];
        const float g1 = Wo[gbase + 8];
        *(v2f*)&wlds[p * PAIR_STRIDE + n * 2] = (v2f){g0, g1};
    }
    __syncthreads();

    const int wave = tid >> 5;
    const int lane = tid & 31;
    const int ln   = lane & 15;   // A row / B,D column sub-index
    const int h    = lane >> 4;   // lane half: K-pair select / D row half

    float bb[4];
#pragma unroll
    for (int t = 0; t < 4; ++t)
        bb[t] = bias[o * N_TOT + t * 16 + ln];

    const int rowBase0 = blockIdx.x * ROWS_PER_BLOCK + wave * 16;
    const float* __restrict__ wlbase = &wlds[h * PAIR_STRIDE + ln * 2];

    for (int it = 0; it < NITER; ++it) {
        const int rowBase = rowBase0 + it * ROWS_PER_ITER;
        // 32-bit per-lane float index (keeps loads in SADDR+voffset form)
        const int xoff = ((rowBase + ln) * 16 + o) * K_TOT + 2 * h;
        const float* __restrict__ xl = x + xoff;

        v8f acc[4];
#pragma unroll
        for (int t = 0; t < 4; ++t)
            acc[t] = (v8f){bb[t], bb[t], bb[t], bb[t],
                           bb[t], bb[t], bb[t], bb[t]};

        // Software pipeline: A fragments prefetched one chunk (K=16) ahead.
        v2f a_buf[4];
#pragma unroll
        for (int j = 0; j < 4; ++j)
            a_buf[j] = nt_load2(xl + 4 * j);

        for (int c = 0; c < 15; ++c) {
            v2f a_nxt[4];
#pragma unroll
            for (int j = 0; j < 4; ++j)
                a_nxt[j] = nt_load2(xl + (c + 1) * 16 + 4 * j);
            chunk4(acc, a_buf, wlbase + (8 * c) * PAIR_STRIDE);
#pragma unroll
            for (int j = 0; j < 4; ++j)
                a_buf[j] = a_nxt[j];
        }
        chunk4(acc, a_buf, wlbase + (8 * 15) * PAIR_STRIDE);

        // Stores: one base + compile-time offsets r*4096B + t*64B (IOFFSET).
        // D layout: VGPR r -> row rowBase + r + 8*h ; lane ln -> column.
        float* __restrict__ obase =
            out + ((size_t)(rowBase + 8 * h) * 16 + o) * N_TOT + ln;
#pragma unroll
        for (int t = 0; t < 4; ++t) {
#pragma unroll
            for (int r = 0; r < 8; ++r)
                __builtin_nontemporal_store(acc[t][r],
                                            obase + r * (16 * N_TOT) + t * 16);
        }
    }
}

extern "C" void kernel_launch(void* const* d_in, const int* in_sizes, int n_in,
                              void* d_out, int out_size, void* d_ws, size_t ws_size,
                              hipStream_t stream) {
    const float* x    = (const float*)d_in[0];   // (16384, 16, 256)
    const float* W    = (const float*)d_in[1];   // (16, 4, 2, 256, 8)
    const float* bias = (const float*)d_in[2];   // (16, 4, 2, 8)
    float* out        = (float*)d_out;           // (16384, 16, 4, 2, 8)

    const int B = 16384, N_OBJ = 16;
    dim3 grid(B / ROWS_PER_BLOCK, N_OBJ);        // (32, 16)
    dim3 block(BDIM);
    const size_t lds_bytes = (size_t)NPAIRS * PAIR_STRIDE * sizeof(float); // 80 KB
    hipLaunchKernelGGL(objdec_wmma_f32, grid, block, lds_bytes, stream,
                       x, W, bias, out);
}